// Pointnet2MSG_48773648614135
// MI455X (gfx1250) — compile-verified
//
#include <hip/hip_runtime.h>
#include <hip/hip_bf16.h>

typedef __attribute__((ext_vector_type(2))) float v2f;
typedef __attribute__((ext_vector_type(8))) float v8f;

#define BQ_THREADS 256
#define BQ_TILE 512

// ---------------------------------------------------------------------------
// Ball query: for each query point, first `ns` indices (ascending) with
// d2 < r2; pad with first hit (or 0 if none). Matches JAX stable-argsort ref.
// ---------------------------------------------------------------------------
__global__ __launch_bounds__(BQ_THREADS)
void ball_query_kernel(const float* __restrict__ xyz,
                       const float* __restrict__ new_xyz,
                       int* __restrict__ idx,
                       int B, int S, int Npts, int ns, float r2) {
    __shared__ float tile[BQ_TILE * 3];
    int sblocks = S / BQ_THREADS;
    int b = blockIdx.x / sblocks;
    int q = (blockIdx.x % sblocks) * BQ_THREADS + threadIdx.x;
    const float* base = xyz + (size_t)b * Npts * 3;
    size_t qg = (size_t)b * S + q;
    float cx = new_xyz[qg * 3 + 0];
    float cy = new_xyz[qg * 3 + 1];
    float cz = new_xyz[qg * 3 + 2];
    int* out = idx + qg * ns;
    int cnt = 0;
    for (int t0 = 0; t0 < Npts; t0 += BQ_TILE) {
        __syncthreads();
        for (int i = threadIdx.x; i < BQ_TILE * 3; i += BQ_THREADS)
            tile[i] = base[(size_t)t0 * 3 + i];
        __syncthreads();
        for (int j = 0; j < BQ_TILE; ++j) {
            float dx = tile[j * 3 + 0] - cx;
            float dy = tile[j * 3 + 1] - cy;
            float dz = tile[j * 3 + 2] - cz;
            float d2 = dx * dx + dy * dy + dz * dz;
            if (d2 < r2 && cnt < ns) { out[cnt] = t0 + j; ++cnt; }
        }
    }
    int first = (cnt > 0) ? out[0] : 0;
    for (int j = cnt; j < ns; ++j) out[j] = first;
}

// ---------------------------------------------------------------------------
// Gather neighbors, subtract center, pad channel 3->4 with zero (WMMA K align)
// ---------------------------------------------------------------------------
__global__ void gather_kernel(const float* __restrict__ xyz,
                              const float* __restrict__ new_xyz,
                              const int* __restrict__ idx,
                              float* __restrict__ x0,
                              int B, int S, int Npts, int ns) {
    int gid = blockIdx.x * blockDim.x + threadIdx.x;
    int total = B * S * ns;
    if (gid >= total) return;
    int q = gid / ns;
    int b = q / S;
    int p = idx[gid];
    const float* src = xyz + ((size_t)b * Npts + p) * 3;
    const float* c   = new_xyz + (size_t)q * 3;
    float* dst = x0 + (size_t)gid * 4;
    dst[0] = src[0] - c[0];
    dst[1] = src[1] - c[1];
    dst[2] = src[2] - c[2];
    dst[3] = 0.0f;
}

// Pad W1 (64x3) -> (64x4) with zero column so the GEMM needs no K guard.
__global__ void pad_w1_kernel(const float* __restrict__ W, float* __restrict__ Wp) {
    int i = threadIdx.x;
    if (i < 64) {
        Wp[i * 4 + 0] = W[i * 3 + 0];
        Wp[i * 4 + 1] = W[i * 3 + 1];
        Wp[i * 4 + 2] = W[i * 3 + 2];
        Wp[i * 4 + 3] = 0.0f;
    }
}

// ---------------------------------------------------------------------------
// WMMA f32 GEMM:  D[M x Ncols] = normA(A[M x K]) * W[Ncols x K]^T
// One wave computes a 32x64 output macro-tile: 2 row-groups x 4 N-tiles,
// 8 accumulators. Per K-step: 6 b64 fragment loads -> 8 WMMAs. Fragment
// loads are software-pipelined one K-step ahead (register double buffering)
// so each load clause is issued before the WMMA burst that hides it.
//   NORM : A-element = relu(a*nscale[k]+nshift[k])  (prev layer BN+ReLU)
//   STORE: store raw GEMM output
//   STATS: per-output-channel sum/sumsq (for train-mode BN)
//   FMAX : v = relu(d*escale[col]+eshift[col]); atomic-max over sample groups
// M multiple of 32, Ncols multiple of 64, K multiple of 4; A and W both have
// K valid (zero-padded) columns; lda/ldw even (b64-aligned fragments).
// ---------------------------------------------------------------------------
template <bool NORM, bool STORE, bool STATS, bool FMAX>
__global__ __launch_bounds__(256)
void gemm_wmma_f32(const float* __restrict__ A, int lda, int K,
                   const float* __restrict__ W, int ldw,
                   const float* __restrict__ nscale, const float* __restrict__ nshift,
                   float* __restrict__ Cst, int ldc,
                   float* __restrict__ sums, float* __restrict__ sumsq,
                   const float* __restrict__ escale, const float* __restrict__ eshift,
                   float* __restrict__ fmax_out, int fmax_ld, int ns, int colOff,
                   int Mrows, int Ncols) {
    int wid  = blockIdx.x * (blockDim.x >> 5) + (threadIdx.x >> 5);
    int lane = threadIdx.x & 31;
    int Ntg = Ncols >> 6;                 // groups of 4 16-wide tiles
    int Mt  = Mrows >> 5;                 // 32-row macro tiles
    if (wid >= Mt * Ntg) return;          // wave-uniform: EXEC stays all-1s
    int mt  = wid / Ntg;
    int ntg = wid - mt * Ntg;
    int m0 = mt << 5, n0 = ntg << 6;
    int half = lane >> 4;                 // 0: K=0,1 / 1: K=2,3 of each step
    int r    = lane & 15;
    int kk   = half << 1;

    const float* Ar0 = A + (size_t)(m0 + r) * lda + kk;
    const float* Ar1 = Ar0 + (size_t)16 * lda;
    const float* Wr0 = W + (size_t)(n0 + r) * ldw + kk;
    const float* Wr1 = Wr0 + (size_t)16 * ldw;
    const float* Wr2 = Wr0 + (size_t)32 * ldw;
    const float* Wr3 = Wr0 + (size_t)48 * ldw;

    auto normA = [&](v2f a, int k) -> v2f {
        if constexpr (NORM) {
            int k0 = k + kk;
            a.x = fmaxf(fmaf(a.x, nscale[k0], nshift[k0]), 0.f);
            a.y = fmaxf(fmaf(a.y, nscale[k0 + 1], nshift[k0 + 1]), 0.f);
        }
        return a;
    };

    v8f acc0a = {0.f,0.f,0.f,0.f,0.f,0.f,0.f,0.f};
    v8f acc1a = acc0a, acc2a = acc0a, acc3a = acc0a;
    v8f acc0b = acc0a, acc1b = acc0a, acc2b = acc0a, acc3b = acc0a;

    // Prologue: fragments for k = 0
    v2f a0 = normA(*(const v2f*)(Ar0), 0);
    v2f a1 = normA(*(const v2f*)(Ar1), 0);
    v2f b0 = *(const v2f*)(Wr0);
    v2f b1 = *(const v2f*)(Wr1);
    v2f b2 = *(const v2f*)(Wr2);
    v2f b3 = *(const v2f*)(Wr3);

    // Pipelined main loop: issue loads for step k, compute step k-4
    for (int k = 4; k < K; k += 4) {
        v2f a0n = normA(*(const v2f*)(Ar0 + k), k);
        v2f a1n = normA(*(const v2f*)(Ar1 + k), k);
        v2f b0n = *(const v2f*)(Wr0 + k);
        v2f b1n = *(const v2f*)(Wr1 + k);
        v2f b2n = *(const v2f*)(Wr2 + k);
        v2f b3n = *(const v2f*)(Wr3 + k);
        acc0a = __builtin_amdgcn_wmma_f32_16x16x4_f32(false, a0, false, b0, (short)0, acc0a, false, false);
        acc1a = __builtin_amdgcn_wmma_f32_16x16x4_f32(false, a0, false, b1, (short)0, acc1a, false, false);
        acc2a = __builtin_amdgcn_wmma_f32_16x16x4_f32(false, a0, false, b2, (short)0, acc2a, false, false);
        acc3a = __builtin_amdgcn_wmma_f32_16x16x4_f32(false, a0, false, b3, (short)0, acc3a, false, false);
        acc0b = __builtin_amdgcn_wmma_f32_16x16x4_f32(false, a1, false, b0, (short)0, acc0b, false, false);
        acc1b = __builtin_amdgcn_wmma_f32_16x16x4_f32(false, a1, false, b1, (short)0, acc1b, false, false);
        acc2b = __builtin_amdgcn_wmma_f32_16x16x4_f32(false, a1, false, b2, (short)0, acc2b, false, false);
        acc3b = __builtin_amdgcn_wmma_f32_16x16x4_f32(false, a1, false, b3, (short)0, acc3b, false, false);
        a0 = a0n; a1 = a1n; b0 = b0n; b1 = b1n; b2 = b2n; b3 = b3n;
    }
    // Final K-step compute
    acc0a = __builtin_amdgcn_wmma_f32_16x16x4_f32(false, a0, false, b0, (short)0, acc0a, false, false);
    acc1a = __builtin_amdgcn_wmma_f32_16x16x4_f32(false, a0, false, b1, (short)0, acc1a, false, false);
    acc2a = __builtin_amdgcn_wmma_f32_16x16x4_f32(false, a0, false, b2, (short)0, acc2a, false, false);
    acc3a = __builtin_amdgcn_wmma_f32_16x16x4_f32(false, a0, false, b3, (short)0, acc3a, false, false);
    acc0b = __builtin_amdgcn_wmma_f32_16x16x4_f32(false, a1, false, b0, (short)0, acc0b, false, false);
    acc1b = __builtin_amdgcn_wmma_f32_16x16x4_f32(false, a1, false, b1, (short)0, acc1b, false, false);
    acc2b = __builtin_amdgcn_wmma_f32_16x16x4_f32(false, a1, false, b2, (short)0, acc2b, false, false);
    acc3b = __builtin_amdgcn_wmma_f32_16x16x4_f32(false, a1, false, b3, (short)0, acc3b, false, false);

    // D layout: acc[i] -> row mbase + i + half*8, col n0 + t*16 + r
    auto epilogue = [&](v8f accA, v8f accB, int t) {
        int col = n0 + t * 16 + r;
        if constexpr (STORE) {
            #pragma unroll
            for (int i = 0; i < 8; ++i) {
                int rowA = m0 + i + half * 8;
                Cst[(size_t)rowA * ldc + col] = accA[i];
                Cst[(size_t)(rowA + 16) * ldc + col] = accB[i];
            }
        }
        if constexpr (STATS) {
            float s = 0.f, s2 = 0.f;
            #pragma unroll
            for (int i = 0; i < 8; ++i) {
                s  += accA[i] + accB[i];
                s2 += accA[i] * accA[i] + accB[i] * accB[i];
            }
            s  += __shfl_xor(s, 16, 32);  // combine the two 8-row halves
            s2 += __shfl_xor(s2, 16, 32);
            if (half == 0) {
                atomicAdd(&sums[col], s);
                atomicAdd(&sumsq[col], s2);
            }
        }
        if constexpr (FMAX) {
            float sc = escale[col], sh = eshift[col];
            #pragma unroll
            for (int i = 0; i < 8; ++i) {
                int rowA = m0 + i + half * 8;
                float vA = fmaxf(fmaf(accA[i], sc, sh), 0.f);
                float vB = fmaxf(fmaf(accB[i], sc, sh), 0.f);
                int qA = rowA / ns;
                int qB = (rowA + 16) / ns;
                __hip_atomic_fetch_max(&fmax_out[(size_t)qA * fmax_ld + colOff + col],
                                       vA, __ATOMIC_RELAXED, __HIP_MEMORY_SCOPE_AGENT);
                __hip_atomic_fetch_max(&fmax_out[(size_t)qB * fmax_ld + colOff + col],
                                       vB, __ATOMIC_RELAXED, __HIP_MEMORY_SCOPE_AGENT);
            }
        }
    };
    epilogue(acc0a, acc0b, 0);
    epilogue(acc1a, acc1b, 1);
    epilogue(acc2a, acc2b, 2);
    epilogue(acc3a, acc3b, 3);
}

// ---------------------------------------------------------------------------
// Per-channel BN train-mode stats -> (scale, shift)
// ---------------------------------------------------------------------------
__global__ void finalize_stats(const float* __restrict__ sums,
                               const float* __restrict__ sumsq,
                               const float* __restrict__ gamma,
                               const float* __restrict__ beta,
                               float* __restrict__ scale,
                               float* __restrict__ shift,
                               float invcnt, int C) {
    int c = blockIdx.x * blockDim.x + threadIdx.x;
    if (c >= C) return;
    float mean = sums[c] * invcnt;
    float var  = sumsq[c] * invcnt - mean * mean;
    float inv  = rsqrtf(var + 1e-5f);
    float sc = gamma[c] * inv;
    scale[c] = sc;
    shift[c] = beta[c] - mean * sc;
}

__global__ void zero_f32(float* __restrict__ p, size_t n) {
    size_t i = (size_t)blockIdx.x * blockDim.x + threadIdx.x;
    if (i < n) p[i] = 0.f;
}

__global__ void bn_relu_out(const float* __restrict__ h,
                            const float* __restrict__ scale,
                            const float* __restrict__ shift,
                            float* __restrict__ out, size_t n, int Cmask) {
    size_t i = (size_t)blockIdx.x * blockDim.x + threadIdx.x;
    if (i >= n) return;
    int c = (int)(i & (size_t)Cmask);
    out[i] = fmaxf(fmaf(h[i], scale[c], shift[c]), 0.f);
}

__global__ void copy_f32(float* __restrict__ dst, const float* __restrict__ src, size_t n) {
    size_t i = (size_t)blockIdx.x * blockDim.x + threadIdx.x;
    if (i < n) dst[i] = src[i];
}

// ---------------------------------------------------------------------------
// Host-side orchestration
// ---------------------------------------------------------------------------
static inline int gemm_blocks(int Mrows, int Ncols) {
    long long waves = (long long)(Mrows >> 5) * (Ncols >> 6);
    return (int)((waves + 7) / 8);
}

extern "C" void kernel_launch(void* const* d_in, const int* in_sizes, int n_in,
                              void* d_out, int out_size, void* d_ws, size_t ws_size,
                              hipStream_t stream) {
    (void)in_sizes; (void)n_in; (void)out_size; (void)ws_size;

    constexpr int B = 2, S = 6144, NP = 16384;
    const int   NSAMP[2] = {16, 32};
    const float R2[2]    = {0.25f, 1.0f};
    const int   COUT[3]  = {64, 128, 1024};

    const float* pointcloud = (const float*)d_in[0];
    const float* clean      = (const float*)d_in[1];
    const float* Wc[2][3], *Gc[2][3], *Bc[2][3];
    for (int s = 0; s < 2; ++s) {
        int base = 2 + s * 9;
        for (int l = 0; l < 3; ++l) {
            Wc[s][l] = (const float*)d_in[base + l];
            Gc[s][l] = (const float*)d_in[base + 3 + l];
            Bc[s][l] = (const float*)d_in[base + 6 + l];
        }
    }
    const float* lin1_w = (const float*)d_in[20];
    const float* bn4_g  = (const float*)d_in[22];
    const float* bn4_b  = (const float*)d_in[23];
    const float* lin2_w = (const float*)d_in[24];
    const float* bn5_g  = (const float*)d_in[26];
    const float* bn5_b  = (const float*)d_in[27];
    // lin1_b / lin2_b (d_in[21], d_in[25]) cancel exactly under train-mode BN.

    // Workspace layout (float units)
    const size_t NMAX = (size_t)B * S * 32;           // 393216 rows max
    float* ws   = (float*)d_ws;
    int*   idxb = (int*)d_ws;                         // NMAX ints
    float* x0   = ws + NMAX;                          // NMAX * 4
    float* x1   = x0 + NMAX * 4;                      // NMAX * 64
    float* x2   = x1 + NMAX * 64;                     // NMAX * 128
    float* feat = x2 + NMAX * 128;                    // 12288 * 2048
    float* st   = feat + (size_t)12288 * 2048;
    float* sums  = st;                                // 1024
    float* sumsq = st + 1024;                         // 1024
    float* scl[5], *shf[5];
    for (int i = 0; i < 5; ++i) { scl[i] = st + 2048 + (size_t)i * 2048; shf[i] = scl[i] + 1024; }
    float* w1pad = st + 2048 + 5 * 2048;              // 256 floats
    float* h1 = x1;                                   // alias: x1/x2 dead after scales
    float* h2 = x2;

    const size_t featN = (size_t)12288 * 2048;
    zero_f32<<<(int)((featN + 255) / 256), 256, 0, stream>>>(feat, featN);

    for (int s = 0; s < 2; ++s) {
        int ns = NSAMP[s];
        int N  = B * S * ns;

        ball_query_kernel<<<B * (S / BQ_THREADS), BQ_THREADS, 0, stream>>>(
            pointcloud, clean, idxb, B, S, NP, ns, R2[s]);
        gather_kernel<<<(N + 255) / 256, 256, 0, stream>>>(
            pointcloud, clean, idxb, x0, B, S, NP, ns);
        pad_w1_kernel<<<1, 64, 0, stream>>>(Wc[s][0], w1pad);

        float invN = 1.0f / (float)N;
        // Layer 1: 3(->4) -> 64, store raw + stats
        zero_f32<<<8, 256, 0, stream>>>(sums, 2048);
        gemm_wmma_f32<false, true, true, false><<<gemm_blocks(N, 64), 256, 0, stream>>>(
            x0, 4, 4, w1pad, 4, nullptr, nullptr, x1, 64, sums, sumsq,
            nullptr, nullptr, nullptr, 0, 0, 0, N, 64);
        finalize_stats<<<1, 256, 0, stream>>>(sums, sumsq, Gc[s][0], Bc[s][0],
                                              scl[0], shf[0], invN, COUT[0]);
        // Layer 2: 64 -> 128, A-load fused BN+ReLU of layer 1
        zero_f32<<<8, 256, 0, stream>>>(sums, 2048);
        gemm_wmma_f32<true, true, true, false><<<gemm_blocks(N, 128), 256, 0, stream>>>(
            x1, 64, 64, Wc[s][1], 64, scl[0], shf[0], x2, 128, sums, sumsq,
            nullptr, nullptr, nullptr, 0, 0, 0, N, 128);
        finalize_stats<<<1, 256, 0, stream>>>(sums, sumsq, Gc[s][1], Bc[s][1],
                                              scl[1], shf[1], invN, COUT[1]);
        // Layer 3 pass A: 128 -> 1024, stats only (avoid 1.6GB store)
        zero_f32<<<8, 256, 0, stream>>>(sums, 2048);
        gemm_wmma_f32<true, false, true, false><<<gemm_blocks(N, 1024), 256, 0, stream>>>(
            x2, 128, 128, Wc[s][2], 128, scl[1], shf[1], nullptr, 0, sums, sumsq,
            nullptr, nullptr, nullptr, 0, 0, 0, N, 1024);
        finalize_stats<<<4, 256, 0, stream>>>(sums, sumsq, Gc[s][2], Bc[s][2],
                                              scl[2], shf[2], invN, COUT[2]);
        // Layer 3 pass B: recompute + fused BN+ReLU+max over samples -> feat
        gemm_wmma_f32<true, false, false, true><<<gemm_blocks(N, 1024), 256, 0, stream>>>(
            x2, 128, 128, Wc[s][2], 128, scl[1], shf[1], nullptr, 0, nullptr, nullptr,
            scl[2], shf[2], feat, 2048, ns, s * 1024, N, 1024);
    }

    const int M = B * S;                  // 12288
    float invM = 1.0f / (float)M;
    // FC1: 2048 -> 1024 (+stats)
    zero_f32<<<8, 256, 0, stream>>>(sums, 2048);
    gemm_wmma_f32<false, true, true, false><<<gemm_blocks(M, 1024), 256, 0, stream>>>(
        feat, 2048, 2048, lin1_w, 2048, nullptr, nullptr, h1, 1024, sums, sumsq,
        nullptr, nullptr, nullptr, 0, 0, 0, M, 1024);
    finalize_stats<<<4, 256, 0, stream>>>(sums, sumsq, bn4_g, bn4_b,
                                          scl[3], shf[3], invM, 1024);
    // FC2: 1024 -> 1024, A-load fused BN+ReLU of FC1 (+stats)
    zero_f32<<<8, 256, 0, stream>>>(sums, 2048);
    gemm_wmma_f32<true, true, true, false><<<gemm_blocks(M, 1024), 256, 0, stream>>>(
        h1, 1024, 1024, lin2_w, 1024, scl[3], shf[3], h2, 1024, sums, sumsq,
        nullptr, nullptr, nullptr, 0, 0, 0, M, 1024);
    finalize_stats<<<4, 256, 0, stream>>>(sums, sumsq, bn5_g, bn5_b,
                                          scl[4], shf[4], invM, 1024);

    // Final BN+ReLU -> d_out, then append clean (tuple return order)
    float* out = (float*)d_out;
    const size_t outN = (size_t)M * 1024;
    bn_relu_out<<<(int)((outN + 255) / 256), 256, 0, stream>>>(
        h2, scl[4], shf[4], out, outN, 1023);
    const size_t cleanN = (size_t)B * S * 3;
    copy_f32<<<(int)((cleanN + 255) / 256), 256, 0, stream>>>(
        out + outN, clean, cleanN);
}